// CounterfactualMultiheadAttention_46153718563227
// MI455X (gfx1250) — compile-verified
//
#include <hip/hip_runtime.h>

#define DM 1024
#define NB 4
#define LL 1024
#define NH 16

typedef __attribute__((ext_vector_type(2))) float v2f;
typedef __attribute__((ext_vector_type(4))) float f4;
typedef __attribute__((ext_vector_type(8))) float v8f;
typedef __attribute__((ext_vector_type(4))) int  i4;

// ---------------------------------------------------------------------------
// Kernel 1: per-batch masked column sum of `value` + count of unmasked keys.
// vbar[b][d] = sum_{k : mask[b][k]==0} value[b][k][d];  cnt[b] = #unmasked.
// grid = (4 col-chunks, 4 batches), block = 256.
// ---------------------------------------------------------------------------
__global__ void k_reduce(const float* __restrict__ value, const int* __restrict__ mask,
                         float* __restrict__ vbar, float* __restrict__ cnt) {
  const int b   = blockIdx.y;
  const int col = blockIdx.x * 256 + threadIdx.x;
  const float* vp = value + (size_t)b * (LL * DM) + col;
  const int*   mp = mask + b * LL;
  float acc = 0.f;
  int c = 0;
  for (int k = 0; k < LL; ++k) {
    const int um = (mp[k] == 0);
    c += um;
    const float v = vp[(size_t)k * DM];
    acc += um ? v : 0.f;
  }
  vbar[b * DM + col] = acc;
  if (blockIdx.x == 0 && threadIdx.x == 0) cnt[b] = (float)c;
}

// ---------------------------------------------------------------------------
// Kernel 2: skinny GEMM  out[m][n] = sum_k (A[m][k]*s_m) * W[n][k] + bias[n],
// M=4, N=K=1024, torch-convention W[out][in] row-major.
// One wave per 16-wide N tile using V_WMMA_F32_16X16X4_F32 (fp32 exact).
// A fragment (16x4 f32): lane<16 -> M=lane, holds K=k0,k0+1; lane>=16 -> K=k0+2,k0+3.
// B fragment (4x16 f32): lane&15 -> N, K-halves split the same way.
// D: VGPR i, lanes 0-15 = row M=i  ->  rows 0..3 live in c[0..3], lanes 0..15.
// grid = 8 blocks x 256 threads = 64 waves = 64 N-tiles.
// ---------------------------------------------------------------------------
__global__ void k_gemm4(const float* __restrict__ A, const float* __restrict__ W,
                        const float* __restrict__ bias, const float* __restrict__ cnt,
                        float* __restrict__ out, int use_scale) {
  const int lane = threadIdx.x & 31;
  const int wave = threadIdx.x >> 5;
  const int n0   = (blockIdx.x * 8 + wave) * 16;
  const int r    = lane & 15;   // A-row index / B-column index within tile
  const int kh   = lane >> 4;   // which K-half this lane supplies

  float s = 1.0f;
  if (use_scale) s = 1.0f / cnt[r & 3];

  const float* arow = A + (r & 3) * DM;             // clamp: rows 4..15 read row r&3, zeroed below
  const float* wrow = W + (size_t)(n0 + r) * DM;
  const float  zmask = (r < 4) ? s : 0.0f;          // folds 1/cnt scale + zero-fill of rows 4..15

  v8f c = {0.f, 0.f, 0.f, 0.f, 0.f, 0.f, 0.f, 0.f};

#if __has_builtin(__builtin_amdgcn_wmma_f32_16x16x4_f32)
#pragma unroll 4
  for (int k0 = 0; k0 < DM; k0 += 4) {
    const int k = k0 + 2 * kh;
    v2f av, bv;
    av.x = arow[k]     * zmask;
    av.y = arow[k + 1] * zmask;
    bv.x = wrow[k];
    bv.y = wrow[k + 1];
    c = __builtin_amdgcn_wmma_f32_16x16x4_f32(false, av, false, bv, (short)0, c,
                                              false, false);
  }
#else
  // Scalar fallback (compile-safe): lane computes out rows directly below.
  if (lane < 16) {
    for (int m = 0; m < 4; ++m) {
      float acc = 0.f;
      const float sm = use_scale ? (1.0f / cnt[m]) : 1.0f;
      const float* am = A + m * DM;
      const float* wn = W + (size_t)(n0 + lane) * DM;
      for (int k = 0; k < DM; ++k) acc += am[k] * sm * wn[k];
      c[m] = acc;
    }
  }
#endif

  if (lane < 16) {
    const float bn = bias[n0 + lane];
#pragma unroll
    for (int m = 0; m < 4; ++m)
      out[m * DM + n0 + lane] = c[m] + bn;
  }
}

// ---------------------------------------------------------------------------
// Kernel 3: output[b][q][:] = outrow[b][:] for all q. 16 MB, 128-bit NT stores.
// ---------------------------------------------------------------------------
__global__ void k_bcast(const float* __restrict__ outrow, float* __restrict__ out) {
  const size_t total = (size_t)NB * LL * DM / 4;  // float4 count = 1,048,576
  const size_t stride = (size_t)gridDim.x * blockDim.x;
  for (size_t i = (size_t)blockIdx.x * blockDim.x + threadIdx.x; i < total; i += stride) {
    const int b  = (int)(i >> 18);  // 1024 rows * 256 f4/row per batch
    const int n4 = (int)(i & 255);
    const f4 v = ((const f4*)outrow)[b * 256 + n4];
    __builtin_nontemporal_store(v, ((f4*)out) + i);
  }
}

// ---------------------------------------------------------------------------
// Kernel 4: attn[b][h][q][k] = mask[b][k] ? 0 : 1/cnt[b]. 256 MB, NT stores.
// ---------------------------------------------------------------------------
__global__ void k_attn(const int* __restrict__ mask, const float* __restrict__ cnt,
                       float* __restrict__ attn) {
  const size_t total = (size_t)NB * NH * LL * LL / 4;  // float4 count = 16,777,216
  const size_t stride = (size_t)gridDim.x * blockDim.x;
  for (size_t i = (size_t)blockIdx.x * blockDim.x + threadIdx.x; i < total; i += stride) {
    const int b  = (int)(i >> 22);  // 16*1024 rows * 256 f4/row per batch
    const int k4 = (int)(i & 255);
    const i4 m = ((const i4*)(mask + b * LL))[k4];
    const float inv = 1.0f / cnt[b];
    f4 v;
    v.x = m.x ? 0.f : inv;
    v.y = m.y ? 0.f : inv;
    v.z = m.z ? 0.f : inv;
    v.w = m.w ? 0.f : inv;
    __builtin_nontemporal_store(v, ((f4*)attn) + i);
  }
}

// ---------------------------------------------------------------------------
// Inputs (setup_inputs order): 0=query 1=key 2=value 3=key_padding_mask
// 4=Wq 5=bq 6=Wk 7=bk 8=Wv 9=bv 10=Wo 11=bo.
// Outputs: output [4,1024,1024] then attn_weights [4,16,1024,1024], fp32.
// Q/K projections provably never affect the outputs.
// ---------------------------------------------------------------------------
extern "C" void kernel_launch(void* const* d_in, const int* in_sizes, int n_in,
                              void* d_out, int out_size, void* d_ws, size_t ws_size,
                              hipStream_t stream) {
  (void)in_sizes; (void)n_in; (void)out_size; (void)ws_size;

  const float* value = (const float*)d_in[2];
  const int*   mask  = (const int*)d_in[3];
  const float* Wv    = (const float*)d_in[8];
  const float* bv    = (const float*)d_in[9];
  const float* Wo    = (const float*)d_in[10];
  const float* bo    = (const float*)d_in[11];

  float* out  = (float*)d_out;
  float* attn = out + (size_t)NB * LL * DM;  // 4,194,304 floats in

  float* w      = (float*)d_ws;
  float* vbar   = w;          // 4096 floats
  float* cnt    = w + 4096;   // 4 floats
  float* ctxbar = w + 4352;   // 4096 floats (16B aligned)
  float* outrow = w + 8448;   // 4096 floats (16B aligned)

  k_reduce<<<dim3(4, 4), 256, 0, stream>>>(value, mask, vbar, cnt);
  k_gemm4 <<<8, 256, 0, stream>>>(vbar,   Wv, bv, cnt, ctxbar, 1);
  k_gemm4 <<<8, 256, 0, stream>>>(ctxbar, Wo, bo, cnt, outrow, 0);
  k_bcast <<<2048, 256, 0, stream>>>(outrow, out);
  k_attn  <<<16384, 256, 0, stream>>>(mask, cnt, attn);
}